// my_GATLayer_9208409882857
// MI455X (gfx1250) — compile-verified
//
#include <hip/hip_runtime.h>

// GAT layer for MI455X (gfx1250, wave32).
// B=128 graphs, N=F=512. Workspace layout (needs ~139 MB):
//   [0, 4MB)        adjacency bitmask (1 bit per (b,n,m))
//   [4MB, 4.25MB)   per-column softmax max   (B*N floats)
//   [4.25MB, 4.5MB) per-column 1/sum         (B*N floats)
//   [4.5MB, ...)    att_raw (B*N*N floats = 128 MB)

#define NB 128
#define NN 512
#define NEG_BIG (-9.0e15f)

typedef float v2f __attribute__((ext_vector_type(2)));
typedef float v8f __attribute__((ext_vector_type(8)));

// ---------------- kernel 0: zero adjacency bitmask ----------------
__global__ void k_zero_mask(unsigned int* __restrict__ mask, int nwords) {
    int i = blockIdx.x * blockDim.x + threadIdx.x;
    if (i < nwords) mask[i] = 0u;
}

// ---------------- kernel 1: scatter edges into bitmask ----------------
__global__ void k_edge_scatter(const int* __restrict__ src,
                               const int* __restrict__ dst,
                               unsigned int* __restrict__ mask, int E) {
    int i = blockIdx.x * blockDim.x + threadIdx.x;
    if (i >= E) return;
    int s = src[i];
    int d = dst[i];
    int g = s >> 9;        // graph id  (s / 512)
    int r = s & 511;       // local row (src)
    int c = d & 511;       // local col (dst)
    unsigned int bit = (unsigned int)((((g << 9) + r) << 9) + c);
    atomicOr(&mask[bit >> 5], 1u << (bit & 31));
}

// ---------------- kernel 2: e = leakyrelu(X * W^T), masked -> att_raw ----------------
// One block = one 64x64 tile of e for one graph. 8 waves; each wave: 16 rows x 32 cols
// (two 16x16 WMMA accumulators sharing one A fragment).
__global__ __launch_bounds__(256)
void k_gemm1(const float* __restrict__ X, const float* __restrict__ W,
             const unsigned int* __restrict__ mask, float* __restrict__ att) {
    __shared__ float ldsX[64][65];   // X tile  [n][k], padded pitch
    __shared__ float ldsW[64][65];   // W tile  [m][k] (read transposed as B)

    const int bid = blockIdx.x;
    const int b   = bid >> 6;
    const int n0  = ((bid >> 3) & 7) << 6;
    const int m0  = (bid & 7) << 6;

    const int tid  = threadIdx.x;
    const int lane = tid & 31;
    const int wave = tid >> 5;
    const int half = lane >> 4;         // K sub-group: {0,1} vs {2,3}
    const int l16  = lane & 15;
    const int R0   = (wave >> 1) << 4;  // wave row   (0,16,32,48)
    const int C0   = (wave & 1) << 5;   // wave col   (0 or 32)

    const int lr  = tid >> 4;           // staging row 0..15
    const int lc4 = (tid & 15) << 2;    // staging col 0..60

    v8f acc0 = {};
    v8f acc1 = {};

    const float* Xrow = X + (size_t)(b * NN + n0) * NN;
    const float* Wrow = W + (size_t)m0 * NN;

    for (int kc = 0; kc < NN; kc += 64) {
#pragma unroll
        for (int rr = 0; rr < 4; ++rr) {
            int r = lr + rr * 16;
            float4 xv = *(const float4*)(Xrow + (size_t)r * NN + kc + lc4);
            ldsX[r][lc4 + 0] = xv.x; ldsX[r][lc4 + 1] = xv.y;
            ldsX[r][lc4 + 2] = xv.z; ldsX[r][lc4 + 3] = xv.w;
            float4 wv = *(const float4*)(Wrow + (size_t)r * NN + kc + lc4);
            ldsW[r][lc4 + 0] = wv.x; ldsW[r][lc4 + 1] = wv.y;
            ldsW[r][lc4 + 2] = wv.z; ldsW[r][lc4 + 3] = wv.w;
        }
        if (kc + 64 < NN)   // hint next tile into cache (global_prefetch_b8)
            __builtin_prefetch(Xrow + (size_t)lr * NN + kc + 64 + lc4, 0, 1);
        __syncthreads();

#pragma unroll
        for (int k0 = 0; k0 < 64; k0 += 4) {
            const int kk = k0 + half * 2;
            v2f a;  a.x  = ldsX[R0 + l16][kk];       a.y  = ldsX[R0 + l16][kk + 1];
            v2f b0; b0.x = ldsW[C0 + l16][kk];       b0.y = ldsW[C0 + l16][kk + 1];
            v2f b1; b1.x = ldsW[C0 + 16 + l16][kk];  b1.y = ldsW[C0 + 16 + l16][kk + 1];
            acc0 = __builtin_amdgcn_wmma_f32_16x16x4_f32(false, a, false, b0,
                                                         (short)0, acc0, false, false);
            acc1 = __builtin_amdgcn_wmma_f32_16x16x4_f32(false, a, false, b1,
                                                         (short)0, acc1, false, false);
        }
        __syncthreads();
    }

    // epilogue: leaky-relu(0.2), adjacency mask, store att_raw
#pragma unroll
    for (int i = 0; i < 8; ++i) {
        int n  = n0 + R0 + half * 8 + i;
        int mA = m0 + C0 + l16;
        unsigned int bitA = (unsigned int)(((b * NN + n) << 9) + mA);
        float va = acc0[i]; va = va > 0.f ? va : 0.2f * va;
        bool  ea = (mask[bitA >> 5] >> (bitA & 31)) & 1u;
        att[(size_t)(b * NN + n) * NN + mA] = ea ? va : NEG_BIG;

        unsigned int bitB = bitA + 16;
        float vb = acc1[i]; vb = vb > 0.f ? vb : 0.2f * vb;
        bool  eb = (mask[bitB >> 5] >> (bitB & 31)) & 1u;
        att[(size_t)(b * NN + n) * NN + mA + 16] = eb ? vb : NEG_BIG;
    }
}

// ---------------- kernel 3: per-column online softmax stats ----------------
// softmax is over axis n for each (b, m). Adjacent threads own adjacent columns
// so every global load is coalesced.
__global__ __launch_bounds__(256)
void k_colstats(const float* __restrict__ att,
                float* __restrict__ cmax, float* __restrict__ cinv) {
    int j = blockIdx.x * 256 + threadIdx.x;     // j = b*N + m
    int b = j >> 9;
    int m = j & 511;
    const float* col = att + (size_t)b * NN * NN + m;
    float mx = -__builtin_huge_valf();
    float s  = 0.f;
    for (int n = 0; n < NN; ++n) {
        float v = col[(size_t)n * NN];
        if (v > mx) { s *= __expf(mx - v); mx = v; }
        s += __expf(v - mx);
    }
    cmax[j] = mx;
    cinv[j] = 1.0f / s;     // empty columns: 512 ones -> uniform, matches jax
}

// ---------------- kernel 4: out = elu(P * X + X), P = normalized softmax ----------------
__global__ __launch_bounds__(256)
void k_gemm2(const float* __restrict__ att, const float* __restrict__ X,
             const float* __restrict__ cmax, const float* __restrict__ cinv,
             float* __restrict__ out) {
    __shared__ float ldsP[64][65];   // P tile [n][m]
    __shared__ float ldsB[64][65];   // X tile [m][f]
    __shared__ float sMax[64];
    __shared__ float sInv[64];

    const int bid = blockIdx.x;
    const int b   = bid >> 6;
    const int n0  = ((bid >> 3) & 7) << 6;
    const int f0  = (bid & 7) << 6;

    const int tid  = threadIdx.x;
    const int lane = tid & 31;
    const int wave = tid >> 5;
    const int half = lane >> 4;
    const int l16  = lane & 15;
    const int R0   = (wave >> 1) << 4;
    const int C0   = (wave & 1) << 5;

    const int lr  = tid >> 4;
    const int lc4 = (tid & 15) << 2;

    v8f acc0 = {};
    v8f acc1 = {};

    for (int mc = 0; mc < NN; mc += 64) {
        if (tid < 64) {
            int j = b * NN + mc + tid;
            sMax[tid] = cmax[j];
            sInv[tid] = cinv[j];
        }
        __syncthreads();
#pragma unroll
        for (int rr = 0; rr < 4; ++rr) {
            int r = lr + rr * 16;
            float4 pv = *(const float4*)(att + (size_t)(b * NN + n0 + r) * NN + mc + lc4);
            ldsP[r][lc4 + 0] = __expf(pv.x - sMax[lc4 + 0]) * sInv[lc4 + 0];
            ldsP[r][lc4 + 1] = __expf(pv.y - sMax[lc4 + 1]) * sInv[lc4 + 1];
            ldsP[r][lc4 + 2] = __expf(pv.z - sMax[lc4 + 2]) * sInv[lc4 + 2];
            ldsP[r][lc4 + 3] = __expf(pv.w - sMax[lc4 + 3]) * sInv[lc4 + 3];
            float4 xv = *(const float4*)(X + (size_t)(b * NN + mc + r) * NN + f0 + lc4);
            ldsB[r][lc4 + 0] = xv.x; ldsB[r][lc4 + 1] = xv.y;
            ldsB[r][lc4 + 2] = xv.z; ldsB[r][lc4 + 3] = xv.w;
        }
        __syncthreads();

#pragma unroll
        for (int k0 = 0; k0 < 64; k0 += 4) {
            const int kk = k0 + half * 2;
            v2f a;  a.x  = ldsP[R0 + l16][kk];  a.y  = ldsP[R0 + l16][kk + 1];
            v2f b0; b0.x = ldsB[kk][C0 + l16];       b0.y = ldsB[kk + 1][C0 + l16];
            v2f b1; b1.x = ldsB[kk][C0 + 16 + l16];  b1.y = ldsB[kk + 1][C0 + 16 + l16];
            acc0 = __builtin_amdgcn_wmma_f32_16x16x4_f32(false, a, false, b0,
                                                         (short)0, acc0, false, false);
            acc1 = __builtin_amdgcn_wmma_f32_16x16x4_f32(false, a, false, b1,
                                                         (short)0, acc1, false, false);
        }
        __syncthreads();
    }

    // epilogue: + X, ELU (alpha=1)
#pragma unroll
    for (int i = 0; i < 8; ++i) {
        int n = n0 + R0 + half * 8 + i;
        size_t idxA = (size_t)(b * NN + n) * NN + f0 + C0 + l16;
        float v = acc0[i] + X[idxA];
        out[idxA] = v > 0.f ? v : (__expf(v) - 1.0f);
        size_t idxB = idxA + 16;
        float w = acc1[i] + X[idxB];
        out[idxB] = w > 0.f ? w : (__expf(w) - 1.0f);
    }
}

extern "C" void kernel_launch(void* const* d_in, const int* in_sizes, int n_in,
                              void* d_out, int out_size, void* d_ws, size_t ws_size,
                              hipStream_t stream) {
    (void)n_in; (void)out_size; (void)ws_size;
    const float* x  = (const float*)d_in[0];
    const float* W  = (const float*)d_in[1];
    const int*   ei = (const int*)d_in[2];   // [2, E] flattened: src then dst
    const int E = in_sizes[2] / 2;

    unsigned char* ws = (unsigned char*)d_ws;
    unsigned int* mask = (unsigned int*)ws;                                  // 4 MB
    float* cmaxp = (float*)(ws + (size_t)(4u << 20));                        // 256 KB
    float* cinvp = (float*)(ws + (size_t)(4u << 20) + (256u << 10));         // 256 KB
    float* att   = (float*)(ws + (size_t)(4u << 20) + (512u << 10));         // 128 MB

    const int nwords = NB * NN * NN / 32;            // 1,048,576
    const int ntiles = NB * (NN / 64) * (NN / 64);   // 8192 blocks

    k_zero_mask   <<<(nwords + 255) / 256, 256, 0, stream>>>(mask, nwords);
    k_edge_scatter<<<(E + 255) / 256,      256, 0, stream>>>(ei, ei + E, mask, E);
    k_gemm1       <<<ntiles,               256, 0, stream>>>(x, W, mask, att);
    k_colstats    <<<NB * NN / 256,        256, 0, stream>>>(att, cmaxp, cinvp);
    k_gemm2       <<<ntiles,               256, 0, stream>>>(att, x, cmaxp, cinvp,
                                                             (float*)d_out);
}